// ResidualBKPConv_2370821947672
// MI455X (gfx1250) — compile-verified
//
#include <hip/hip_runtime.h>

// ---- static problem config (mirrors reference) ----
#define B_     4
#define N_     8192
#define CIN    128
#define COUT   256
#define M_     2048
#define KPTS   16
#define DMID   64
#define MAXNN  64
#define RADIUS_ 0.1f
#define R2_    (RADIUS_ * RADIUS_)
#define KPEXT_ RADIUS_

typedef __attribute__((ext_vector_type(2))) float v2f;
typedef __attribute__((ext_vector_type(8))) float v8f;

static __device__ __forceinline__ v8f wmma_f32(v2f a, v2f b, v8f c) {
  // D = A(16x4, f32) x B(4x16, f32) + C(16x16, f32)
  return __builtin_amdgcn_wmma_f32_16x16x4_f32(false, a, false, b, (short)0, c,
                                               false, false);
}

// ---------------------------------------------------------------------------
// Kernel 1: farthest point sampling, one block per cloud.
// Points + running min-dist live in registers (16 per thread); argmax via LDS
// tree reduction with first-index tie-break to match jnp.argmax.
// ---------------------------------------------------------------------------
__global__ void __launch_bounds__(512) fps_kernel(const float* __restrict__ pos,
                                                  int* __restrict__ idx) {
  const int b = blockIdx.x;
  const int t = threadIdx.x;
  const float* p = pos + (size_t)b * N_ * 3;
  constexpr int PPT = N_ / 512;  // 16 points per thread

  float px[PPT], py[PPT], pz[PPT], mind[PPT];
  const float x0 = p[0], y0 = p[1], z0 = p[2];
#pragma unroll
  for (int i = 0; i < PPT; ++i) {
    const int j = t + i * 512;
    px[i] = p[j * 3 + 0];
    py[i] = p[j * 3 + 1];
    pz[i] = p[j * 3 + 2];
    const float dx = px[i] - x0, dy = py[i] - y0, dz = pz[i] - z0;
    mind[i] = dx * dx + dy * dy + dz * dz;
  }
  if (t == 0) idx[b * M_ + 0] = 0;

  __shared__ float sval[512];
  __shared__ int sidx[512];

  for (int m = 1; m < M_; ++m) {
    float bv = -1.0f;
    int bi = 0;
#pragma unroll
    for (int i = 0; i < PPT; ++i) {
      const int j = t + i * 512;
      if (mind[i] > bv) { bv = mind[i]; bi = j; }  // j ascending -> first max
    }
    sval[t] = bv;
    sidx[t] = bi;
    __syncthreads();
    for (int s = 256; s > 0; s >>= 1) {
      if (t < s) {
        const float ov = sval[t + s];
        const int oi = sidx[t + s];
        if (ov > sval[t] || (ov == sval[t] && oi < sidx[t])) {
          sval[t] = ov;
          sidx[t] = oi;
        }
      }
      __syncthreads();
    }
    const int best = sidx[0];
    if (t == 0) idx[b * M_ + m] = best;
    const float bx = p[best * 3 + 0], by = p[best * 3 + 1], bz = p[best * 3 + 2];
#pragma unroll
    for (int i = 0; i < PPT; ++i) {
      const float dx = px[i] - bx, dy = py[i] - by, dz = pz[i] - bz;
      mind[i] = fminf(mind[i], dx * dx + dy * dy + dz * dz);
    }
    __syncthreads();  // protect sval/sidx for next iteration
  }
}

// ---------------------------------------------------------------------------
// Kernel 2: x_pre = x @ W_pre + b_pre  ([B*N,128] @ [128,64]) via WMMA f32.
// One wave per 16x16 output tile; k-loop of 32 WMMA steps.
// ---------------------------------------------------------------------------
__global__ void __launch_bounds__(256)
pre_mlp_kernel(const float* __restrict__ x, const float* __restrict__ Wp,
               const float* __restrict__ bp, float* __restrict__ xpre) {
  const int lane = threadIdx.x & 31;
  const int wave = threadIdx.x >> 5;
  const int wg = blockIdx.x * 8 + wave;  // 8192 tiles total
  const int rt = wg >> 2;                // row tile (2048)
  const int ctile = wg & 3;              // col tile (4)
  const int l16 = lane & 15;
  const int khalf = (lane >> 4) << 1;    // 0 or 2
  const int arow = rt * 16 + l16;
  const int col = ctile * 16 + l16;

  v8f acc = {0.f, 0.f, 0.f, 0.f, 0.f, 0.f, 0.f, 0.f};
#pragma unroll 4
  for (int ks = 0; ks < CIN / 4; ++ks) {
    const int k0 = ks * 4 + khalf;
    v2f a, bf;
    a.x = x[(size_t)arow * CIN + k0];
    a.y = x[(size_t)arow * CIN + k0 + 1];
    bf.x = Wp[(size_t)k0 * DMID + col];
    bf.y = Wp[(size_t)(k0 + 1) * DMID + col];
    acc = wmma_f32(a, bf, acc);
  }
  const float bias = bp[col];
#pragma unroll
  for (int r = 0; r < 8; ++r) {
    const int orow = rt * 16 + r + ((lane >> 4) << 3);
    xpre[(size_t)orow * DMID + col] = acc[r] + bias;
  }
}

// ---------------------------------------------------------------------------
// Kernel 3: radius neighbor search, one wave per center.
// Ballot-compacted append of in-ball point indices (cap 64, pad -1).
// Also writes out_pos (sampled centers).
// ---------------------------------------------------------------------------
__global__ void __launch_bounds__(256)
neighbor_kernel(const float* __restrict__ pos, const int* __restrict__ idx,
                int* __restrict__ nn, float* __restrict__ out_pos) {
  const int lane = threadIdx.x & 31;
  const int wave = threadIdx.x >> 5;
  const int g = blockIdx.x * 8 + wave;  // global center id in [0, B*M)
  const int b = g / M_;
  const int ci = idx[g];
  const float* p = pos + (size_t)b * N_ * 3;
  const float cx = p[ci * 3 + 0], cy = p[ci * 3 + 1], cz = p[ci * 3 + 2];
  if (lane == 0) {
    out_pos[g * 3 + 0] = cx;
    out_pos[g * 3 + 1] = cy;
    out_pos[g * 3 + 2] = cz;
  }
  int cnt = 0;
  for (int j0 = 0; j0 < N_; j0 += 32) {
    const int j = j0 + lane;
    const float dx = p[j * 3 + 0] - cx, dy = p[j * 3 + 1] - cy,
                dz = p[j * 3 + 2] - cz;
    const bool in = (dx * dx + dy * dy + dz * dz) <= R2_;
    const unsigned mask = (unsigned)__ballot(in);
    const int mypos = cnt + __popc(mask & ((1u << lane) - 1u));
    if (in && mypos < MAXNN) nn[(size_t)g * MAXNN + mypos] = j;
    cnt += __popc(mask);
    if (cnt >= MAXNN) break;  // cnt is wave-uniform
  }
  if (cnt > MAXNN) cnt = MAXNN;
  for (int e = cnt + lane; e < MAXNN; e += 32) nn[(size_t)g * MAXNN + e] = -1;
}

// ---------------------------------------------------------------------------
// Kernel 4: fused KP-conv + post MLP + shortcut. 8 centers per block
// (one per wave), 256 threads.
//   stage A: s[16x64] = infl^T[16x64] @ x_j[64x64]        (64 WMMA / wave)
//   stage B: mid[16x64] = S[16x1024] @ kp_W[1024x64]      (128 WMMA / wave,
//            K split across 8 waves, LDS sum of partials)
//   stage C: out[8x256] = mid[8x64]@W_post + x_c[8x128]@W_sc + biases
//            (batched 16-row WMMA tiles, rows 8..15 zero-padded)
// ---------------------------------------------------------------------------
__global__ void __launch_bounds__(256)
conv_kernel(const float* __restrict__ x, const float* __restrict__ pos,
            const float* __restrict__ kp_pos, const float* __restrict__ kp_W,
            const float* __restrict__ W_post, const float* __restrict__ b_post,
            const float* __restrict__ W_sc, const float* __restrict__ b_sc,
            const int* __restrict__ idx, const int* __restrict__ nn,
            const float* __restrict__ xpre, float* __restrict__ out_feat) {
  const int tid = threadIdx.x;
  const int lane = tid & 31;
  const int wave = tid >> 5;
  const int g = blockIdx.x * 8 + wave;
  const int b = g / M_;
  const int ci = idx[g];
  const int l16 = lane & 15;
  const int khalf = (lane >> 4) << 1;  // 0 or 2

  __shared__ float buf[8][KPTS * DMID];  // infl[e][k] -> s[k][d] -> partials
  __shared__ int nns[8][MAXNN];
  __shared__ float mid[16 * DMID];  // S @ kp_W result (rows 8..15 exact zero)
  __shared__ float xs[16 * CIN];    // shortcut rows, rows 8..15 zero
  __shared__ float kpp[KPTS * 3];

  for (int i = tid; i < 8 * CIN; i += 256) xs[8 * CIN + i] = 0.f;
  if (tid < KPTS * 3) kpp[tid] = kp_pos[tid];
  __syncthreads();

  // neighbor list + center position
  nns[wave][lane] = nn[(size_t)g * MAXNN + lane];
  nns[wave][lane + 32] = nn[(size_t)g * MAXNN + lane + 32];
  const float* p = pos + (size_t)b * N_ * 3;
  const float cx = p[ci * 3 + 0], cy = p[ci * 3 + 1], cz = p[ci * 3 + 2];

  // kernel-point influences -> buf[wave] as infl[e][k]
#pragma unroll
  for (int h = 0; h < 2; ++h) {
    const int e = lane + h * 32;
    const int j = nns[wave][e];
    const float valid = (j >= 0) ? 1.f : 0.f;
    const int jj = (j >= 0) ? j : 0;
    const float rx = p[jj * 3 + 0] - cx, ry = p[jj * 3 + 1] - cy,
                rz = p[jj * 3 + 2] - cz;
#pragma unroll
    for (int k = 0; k < KPTS; ++k) {
      const float dx = rx - kpp[k * 3 + 0], dy = ry - kpp[k * 3 + 1],
                  dz = rz - kpp[k * 3 + 2];
      const float dist = sqrtf(dx * dx + dy * dy + dz * dz + 1e-12f);
      buf[wave][e * KPTS + k] =
          fmaxf(1.f - dist * (1.f / KPEXT_), 0.f) * valid;
    }
  }

  // ---- Stage A: s = infl^T @ x_j via WMMA (reduce over e = 64) ----
  v8f sA[4];
#pragma unroll
  for (int nt = 0; nt < 4; ++nt)
    sA[nt] = (v8f){0.f, 0.f, 0.f, 0.f, 0.f, 0.f, 0.f, 0.f};
  const float* xpb = xpre + (size_t)b * N_ * DMID;
  for (int es = 0; es < MAXNN / 4; ++es) {
    const int e0 = es * 4 + khalf;
    v2f a;
    a.x = buf[wave][e0 * KPTS + l16];        // infl[e0][k=l16]
    a.y = buf[wave][(e0 + 1) * KPTS + l16];  // infl[e0+1][k=l16]
    int j0 = nns[wave][e0];
    j0 = (j0 >= 0) ? j0 : 0;
    int j1 = nns[wave][e0 + 1];
    j1 = (j1 >= 0) ? j1 : 0;
    // prefetch next iteration's gather rows (global_prefetch_b8)
    {
      const int esn = (es + 1 < MAXNN / 4) ? (es + 1) : es;
      const int en0 = esn * 4 + khalf;
      int jn = nns[wave][en0];
      jn = (jn >= 0) ? jn : 0;
      __builtin_prefetch(&xpb[(size_t)jn * DMID], 0, 1);
    }
#pragma unroll
    for (int nt = 0; nt < 4; ++nt) {
      v2f bf;
      bf.x = xpb[(size_t)j0 * DMID + nt * 16 + l16];
      bf.y = xpb[(size_t)j1 * DMID + nt * 16 + l16];
      sA[nt] = wmma_f32(a, bf, sA[nt]);
    }
  }
  // spill s[k][d] into the same LDS buffer (wave-private, DS in-order)
#pragma unroll
  for (int nt = 0; nt < 4; ++nt)
#pragma unroll
    for (int r = 0; r < 8; ++r) {
      const int krow = r + ((lane >> 4) << 3);
      buf[wave][krow * DMID + nt * 16 + l16] = sA[nt][r];
    }
  __syncthreads();  // all s matrices visible block-wide

  // ---- Stage B: mid = S[16x1024] @ kp_W[1024x64] via WMMA ----
  // Wave handles K-slice [wave*128, wave*128+128), all 4 col tiles.
  // A rows: S[m][kd] = buf[m][kd] for m<8, zero for m>=8 (padding rows).
  v8f mb[4];
#pragma unroll
  for (int nt = 0; nt < 4; ++nt)
    mb[nt] = (v8f){0.f, 0.f, 0.f, 0.f, 0.f, 0.f, 0.f, 0.f};
  {
    const int mrow = l16 & 7;             // clamped LDS row (valid address)
    const float rowv = (l16 < 8) ? 1.f : 0.f;
    const int kbase = wave * 128;
    for (int ks = 0; ks < 32; ++ks) {
      const int k0 = kbase + ks * 4 + khalf;
      v2f a;
      a.x = buf[mrow][k0] * rowv;
      a.y = buf[mrow][k0 + 1] * rowv;
#pragma unroll
      for (int nt = 0; nt < 4; ++nt) {
        v2f bf;
        bf.x = kp_W[(size_t)k0 * DMID + nt * 16 + l16];
        bf.y = kp_W[(size_t)(k0 + 1) * DMID + nt * 16 + l16];
        mb[nt] = wmma_f32(a, bf, mb[nt]);
      }
    }
  }
  __syncthreads();  // all reads of S complete before overwriting buf
  // write this wave's partial [16x64] into buf[wave]
#pragma unroll
  for (int nt = 0; nt < 4; ++nt)
#pragma unroll
    for (int r = 0; r < 8; ++r) {
      const int m = r + ((lane >> 4) << 3);
      buf[wave][m * DMID + nt * 16 + l16] = mb[nt][r];
    }
  // shortcut features for this center
#pragma unroll
  for (int q = 0; q < 4; ++q)
    xs[wave * CIN + lane + q * 32] =
        x[((size_t)b * N_ + ci) * CIN + lane + q * 32];
  __syncthreads();
  // reduce the 8 K-partials -> mid[16x64]
  for (int i = tid; i < 16 * DMID; i += 256) {
    float v = 0.f;
#pragma unroll
    for (int w = 0; w < 8; ++w) v += buf[w][i];
    mid[i] = v;
  }
  __syncthreads();

  // ---- Stage C: batched post + shortcut GEMMs, 2 col tiles per wave ----
#pragma unroll
  for (int t2 = 0; t2 < 2; ++t2) {
    const int ct = wave * 2 + t2;
    const int col = ct * 16 + l16;
    v8f acc = {0.f, 0.f, 0.f, 0.f, 0.f, 0.f, 0.f, 0.f};
    for (int ks = 0; ks < DMID / 4; ++ks) {  // mid @ W_post
      const int k0 = ks * 4 + khalf;
      v2f a, bf;
      a.x = mid[l16 * DMID + k0];
      a.y = mid[l16 * DMID + k0 + 1];
      bf.x = W_post[(size_t)k0 * COUT + col];
      bf.y = W_post[(size_t)(k0 + 1) * COUT + col];
      acc = wmma_f32(a, bf, acc);
    }
    for (int ks = 0; ks < CIN / 4; ++ks) {  // xs @ W_sc
      const int k0 = ks * 4 + khalf;
      v2f a, bf;
      a.x = xs[l16 * CIN + k0];
      a.y = xs[l16 * CIN + k0 + 1];
      bf.x = W_sc[(size_t)k0 * COUT + col];
      bf.y = W_sc[(size_t)(k0 + 1) * COUT + col];
      acc = wmma_f32(a, bf, acc);
    }
    const float bias = b_post[col] + b_sc[col];
    if (lane < 16) {  // rows 0..7 (valid centers) live in lanes 0..15
#pragma unroll
      for (int r = 0; r < 8; ++r) {
        const int gr = blockIdx.x * 8 + r;
        out_feat[(size_t)gr * COUT + col] = acc[r] + bias;
      }
    }
  }
}

// ---------------------------------------------------------------------------
extern "C" void kernel_launch(void* const* d_in, const int* in_sizes, int n_in,
                              void* d_out, int out_size, void* d_ws,
                              size_t ws_size, hipStream_t stream) {
  const float* x = (const float*)d_in[0];
  const float* pos = (const float*)d_in[1];
  const float* W_pre = (const float*)d_in[2];
  const float* b_pre = (const float*)d_in[3];
  const float* kp_pos = (const float*)d_in[4];
  const float* kp_W = (const float*)d_in[5];
  const float* W_post = (const float*)d_in[6];
  const float* b_post = (const float*)d_in[7];
  const float* W_sc = (const float*)d_in[8];
  const float* b_sc = (const float*)d_in[9];

  float* out_feat = (float*)d_out;                     // [B,M,COUT]
  float* out_pos = out_feat + (size_t)B_ * M_ * COUT;  // [B,M,3]

  // workspace carve-out (~12.7 MB total)
  char* ws = (char*)d_ws;
  int* idx = (int*)ws;                // B*M ints
  int* nn = (int*)(ws + (64 << 10));  // B*M*64 ints (2MB)
  float* xpre =
      (float*)(ws + (64 << 10) + (2u << 21) + (64 << 10));  // B*N*64 f32

  fps_kernel<<<B_, 512, 0, stream>>>(pos, idx);
  pre_mlp_kernel<<<(B_ * N_ / 16) * (DMID / 16) / 8, 256, 0, stream>>>(
      x, W_pre, b_pre, xpre);
  neighbor_kernel<<<B_ * M_ / 8, 256, 0, stream>>>(pos, idx, nn, out_pos);
  conv_kernel<<<B_ * M_ / 8, 256, 0, stream>>>(x, pos, kp_pos, kp_W, W_post,
                                               b_post, W_sc, b_sc, idx, nn,
                                               xpre, out_feat);
}